// Ricker_42210938585243
// MI455X (gfx1250) — compile-verified
//
#include <hip/hip_runtime.h>
#include <cstdint>

// ---------------------------------------------------------------------------
// Ricker-map scan, MI455X (gfx1250).
// Serial nonlinear recurrence -> latency bound. Critical path per step is
// exactly one v_exp_f32 + one v_fma_f32 (log2-domain state). All memory is
// hidden behind gfx1250 async global->LDS double buffering (ASYNCcnt).
// ---------------------------------------------------------------------------

#define AS1 __attribute__((address_space(1)))
#define AS3 __attribute__((address_space(3)))

typedef int rk_v4i __attribute__((vector_size(4 * sizeof(int))));

#if __has_builtin(__builtin_amdgcn_global_load_async_to_lds_b128) && \
    __has_builtin(__builtin_amdgcn_s_wait_asynccnt)
#define RK_HAVE_ASYNC 1
#else
#define RK_HAVE_ASYNC 0
#endif

#define RK_TILE 2048  // floats per LDS buffer (8 KB) -> 16 async b128 ops/tile

__device__ __forceinline__ float rk_exp2(float x) {
#if __has_builtin(__builtin_amdgcn_exp2f)
  return __builtin_amdgcn_exp2f(x);   // v_exp_f32 (computes 2^x)
#else
  return exp2f(x);
#endif
}

__device__ __forceinline__ float rk_log2(float x) {
#if __has_builtin(__builtin_amdgcn_logf)
  return __builtin_amdgcn_logf(x);    // v_log_f32 (computes log2 x)
#else
  return log2f(x);
#endif
}

__global__ __launch_bounds__(32)
void Ricker_scan_gfx1250(const float* __restrict__ N0,
                         const float* __restrict__ Temp,
                         const float* __restrict__ P,
                         float* __restrict__ out, int Tlen) {
  __shared__ float lds[2][RK_TILE];
  const int lane = (int)threadIdx.x;
  const long nsteps = (long)Tlen - 1;  // steps consume Temp[0..nsteps-1]

  // Fold log2(e) into all coefficients: state M = log2(n).
  // M_{i+1} = M_i + (A + B*t + C*t^2) - D*n_i ;  n_i = exp2(M_i)
  const float alpha = P[0], beta = P[1], bx = P[2], cx = P[3];
  const float L2E = 1.4426950408889634f;
  const float A = L2E * alpha;
  const float B = A * bx;
  const float C = A * cx;
  const float D = A * beta;

  float n = N0[0];
  float M = rk_log2(n);

  if (Tlen < 64) {  // tiny-problem scalar path (no LDS/barriers)
    if (lane == 0 && nsteps >= 0) {
      for (long i = 0; i < nsteps; ++i) {
        const float tt = Temp[i];
        const float g = fmaf(tt, fmaf(C, tt, B), A);
        out[i] = n;
        M = fmaf(-D, n, M + g);
        n = rk_exp2(M);
      }
      out[nsteps] = n;
    }
    return;
  }

  const int ntiles = (int)((nsteps + RK_TILE - 1) / RK_TILE);

  // Whole-wave cooperative stage of one tile of Temp into LDS buffer (t&1).
  // Async path: each global_load_async_to_lds_b128 moves 32 lanes x 16B.
  auto issue_tile = [&](int t) {
    const long base = (long)t * RK_TILE;
#pragma unroll
    for (int s = 0; s < RK_TILE / 128; ++s) {
      long gi = base + (long)s * 128 + (long)lane * 4;
      const long gmax = (long)Tlen - 4;  // clamp: stay in-bounds (clamped chunks hold shifted data)
      if (gi > gmax) gi = gmax;
      const float* gp = Temp + gi;
      float* lp = &lds[t & 1][s * 128 + lane * 4];
#if RK_HAVE_ASYNC
      __builtin_amdgcn_global_load_async_to_lds_b128(
          (AS1 rk_v4i*)(void*)gp, (AS3 rk_v4i*)(void*)lp, 0, 0);
#else
      const float4 v = *(const float4*)gp;
      *(float4*)lp = v;
#endif
    }
  };

  issue_tile(0);
  for (int t = 0; t < ntiles; ++t) {
    if (t + 1 < ntiles) {
      issue_tile(t + 1);  // prefetch next tile into the other buffer
#if RK_HAVE_ASYNC
      // 16 loads just issued for tile t+1; async loads retire in order, so
      // ASYNCcnt <= 16 implies tile t's loads are complete.
      __builtin_amdgcn_s_wait_asynccnt(RK_TILE / 128);
#endif
    } else {
#if RK_HAVE_ASYNC
      __builtin_amdgcn_s_wait_asynccnt(0);
#endif
    }
    __syncthreads();  // single-wave WG: compiler ordering + DS visibility
    asm volatile("" ::: "memory");

    const long base = (long)t * RK_TILE;
    const long rem = nsteps - base;
    const int count = rem > RK_TILE ? RK_TILE : (int)rem;

    if (lane == 0) {
      const float4* __restrict__ lb4 = (const float4*)&lds[t & 1][0];
      float4* __restrict__ ob4 = (float4*)(out + base);  // base%RK_TILE==0 -> 16B aligned
      const int full = count >> 2;
      for (int q = 0; q < full; ++q) {
        const float4 t4 = lb4[q];
        // Forcing terms: entirely off the serial critical path.
        const float g0 = fmaf(t4.x, fmaf(C, t4.x, B), A);
        const float g1 = fmaf(t4.y, fmaf(C, t4.y, B), A);
        const float g2 = fmaf(t4.z, fmaf(C, t4.z, B), A);
        const float g3 = fmaf(t4.w, fmaf(C, t4.w, B), A);
        float4 o;
        // Critical path per step: fma -> v_exp_f32.
        o.x = n; M = fmaf(-D, n, M + g0); n = rk_exp2(M);
        o.y = n; M = fmaf(-D, n, M + g1); n = rk_exp2(M);
        o.z = n; M = fmaf(-D, n, M + g2); n = rk_exp2(M);
        o.w = n; M = fmaf(-D, n, M + g3); n = rk_exp2(M);
        ob4[q] = o;
      }
      const float* lb = &lds[t & 1][0];
      for (int j = full << 2; j < count; ++j) {  // tail (last tile only)
        const float tt = lb[j];
        const float g = fmaf(tt, fmaf(C, tt, B), A);
        out[base + j] = n;
        M = fmaf(-D, n, M + g);
        n = rk_exp2(M);
      }
    }
    asm volatile("" ::: "memory");
    __syncthreads();  // lane0's DS reads retired before buffer reuse
  }
  if (lane == 0) out[nsteps] = n;  // out[T-1] = final state
}

extern "C" void kernel_launch(void* const* d_in, const int* in_sizes, int n_in,
                              void* d_out, int out_size, void* d_ws, size_t ws_size,
                              hipStream_t stream) {
  (void)n_in; (void)out_size; (void)d_ws; (void)ws_size;
  const float* N0   = (const float*)d_in[0];
  const float* Temp = (const float*)d_in[1];
  const float* P    = (const float*)d_in[2];
  float* out = (float*)d_out;
  const int T = in_sizes[1];
  hipLaunchKernelGGL(Ricker_scan_gfx1250, dim3(1), dim3(32), 0, stream,
                     N0, Temp, P, out, T);
}